// GatDot_rel_44727789421069
// MI455X (gfx1250) — compile-verified
//
#include <hip/hip_runtime.h>
#include <hip/hip_bf16.h>
#include <math.h>

typedef __attribute__((ext_vector_type(2))) float v2f;
typedef __attribute__((ext_vector_type(8))) float v8f;

#define BB 1024
#define NN 512
#define DD 256

// ---------------------------------------------------------------------------
// Kernel 1: fp32 WMMA GEMM  C[M,N] = A[M,K] * B (+ bias)
//   BT=true : B element [k,n] = Bm[n*ldb + k]   (i.e. C = A * Bmᵀ)
//   BT=false: B element [k,n] = Bm[k*ldb + n]   (i.e. C = A * Bm)
// One wave (32 threads) per 16x16 C tile, K-loop in steps of 4 using
// v_wmma_f32_16x16x4_f32 (full fp32 precision, matches reference).
// ---------------------------------------------------------------------------
template <bool BT, bool BIAS>
__launch_bounds__(32)
__global__ void wmma_gemm_f32(const float* __restrict__ A,
                              const float* __restrict__ Bm,
                              const float* __restrict__ bias,
                              float* __restrict__ C,
                              int M, int N, int K, int lda, int ldb, int ldc) {
  const int lane = threadIdx.x;            // 0..31, full wave (EXEC all ones)
  const int n0 = blockIdx.x * 16;
  const int m0 = blockIdx.y * 16;
  const int r  = lane & 15;                // M (for A) / N (for B) index
  const int kh = (lane >> 4) * 2;          // K sub-offset: 0 or 2

  v8f acc = {};
  for (int k0 = 0; k0 < K; k0 += 4) {
    // A fragment: lane holds A[m0+r, k0+kh+{0,1}]  (contiguous b64)
    const float* ap = A + (size_t)(m0 + r) * lda + (k0 + kh);
    v2f a; a.x = ap[0]; a.y = ap[1];
    // B fragment: lane holds B[k0+kh+{0,1}, n0+r]
    v2f b;
    if (BT) {
      const float* bp = Bm + (size_t)(n0 + r) * ldb + (k0 + kh);
      b.x = bp[0]; b.y = bp[1];            // contiguous along Bm row
    } else {
      const float* bp = Bm + (size_t)(k0 + kh) * ldb + (n0 + r);
      b.x = bp[0]; b.y = bp[ldb];
    }
    acc = __builtin_amdgcn_wmma_f32_16x16x4_f32(
        /*neg_a=*/false, a, /*neg_b=*/false, b,
        /*c_mod=*/(short)0, acc, /*reuse_a=*/false, /*reuse_b=*/false);
  }

  // D layout: VGPR i, this lane -> C[m0 + 8*(lane/16) + i, n0 + (lane%16)]
  const int mbase = m0 + (lane >> 4) * 8;
  const int nc    = n0 + r;
  const float bv  = BIAS ? bias[nc] : 0.0f;
#pragma unroll
  for (int i = 0; i < 8; ++i)
    C[(size_t)(mbase + i) * ldc + nc] = acc[i] + bv;
}

// ---------------------------------------------------------------------------
// Kernel 2: scalar precompute  y_j = rel_emb[j,:]·W3[0,:] + b3   (j = 0,1)
// ---------------------------------------------------------------------------
__global__ void y_precompute(const float* __restrict__ rel_emb,
                             const float* __restrict__ W3,
                             const float* __restrict__ b3,
                             float* __restrict__ yout) {
  __shared__ float s0[DD], s1[DD];
  const int d = threadIdx.x;
  const float w = W3[d];
  s0[d] = rel_emb[d] * w;
  s1[d] = rel_emb[DD + d] * w;
  __syncthreads();
  for (int off = DD / 2; off > 0; off >>= 1) {
    if (d < off) { s0[d] += s0[d + off]; s1[d] += s1[d + off]; }
    __syncthreads();
  }
  if (d == 0) { yout[0] = s0[0] + b3[0]; yout[1] = s1[0] + b3[0]; }
}

// ---------------------------------------------------------------------------
// Kernel 3: c[b] = q[b,:]·b2   (b2 bias of the k-projection, folded scalar)
// ---------------------------------------------------------------------------
__global__ void qdotb2(const float* __restrict__ q,
                       const float* __restrict__ b2,
                       float* __restrict__ c) {
  __shared__ float s[DD];
  const int b = blockIdx.x, t = threadIdx.x;
  s[t] = q[(size_t)b * DD + t] * b2[t];
  __syncthreads();
  for (int off = DD / 2; off > 0; off >>= 1) {
    if (t < off) s[t] += s[t + off];
    __syncthreads();
  }
  if (t == 0) c[b] = s[0];
}

// ---------------------------------------------------------------------------
// Kernel 4: the bandwidth-bound core. One workgroup (512 thr = 16 waves)
// per batch b. Streams K[b] (512KB) and V[b] (512KB) exactly once.
//   alpha[n] = K[b,n,:]·p[b,:] + c[b] + y[s_mask] - (1-adj)*1e30
//   w = softmax(alpha);  attn_sum[d] = sum_n w[n]*V[b,n,d]
// ---------------------------------------------------------------------------
__launch_bounds__(512)
__global__ void gat_attention(const float* __restrict__ K,
                              const float* __restrict__ V,
                              const int* __restrict__ adj,
                              const int* __restrict__ s_mask,
                              const float* __restrict__ p,   // B x D
                              const float* __restrict__ yv,  // 2
                              const float* __restrict__ cv,  // B
                              float* __restrict__ out_w,     // B x N
                              float* __restrict__ out_s) {   // B x D
  const int b    = blockIdx.x;
  const int tid  = threadIdx.x;   // 0..511
  const int lane = tid & 31;
  const int wave = tid >> 5;      // 0..15

  __shared__ float sp[DD];        // p[b,:]
  __shared__ float sw[NN];        // alpha -> softmax weights -> partials
  __shared__ float wred[16];      // cross-wave reduction

  if (tid < DD) sp[tid] = p[(size_t)b * DD + tid];
  __syncthreads();

  const float y0 = yv[0], y1 = yv[1], cb = cv[b];
  const float* Kb = K + (size_t)b * NN * DD;

  // p chunk for this lane (8 floats, lane*8 .. lane*8+7)
  float pl[8];
#pragma unroll
  for (int i = 0; i < 8; ++i) pl[i] = sp[lane * 8 + i];

  // wave-cooperative row dots: wave handles rows [wave*32, wave*32+32)
  for (int nn = 0; nn < 32; ++nn) {
    const int n = wave * 32 + nn;
    const float* kr = Kb + (size_t)n * DD + lane * 8;   // 32B-aligned
    const float4 k0 = *(const float4*)(kr);
    const float4 k1 = *(const float4*)(kr + 4);
    float acc = k0.x * pl[0] + k0.y * pl[1] + k0.z * pl[2] + k0.w * pl[3] +
                k1.x * pl[4] + k1.y * pl[5] + k1.z * pl[6] + k1.w * pl[7];
#pragma unroll
    for (int off = 16; off > 0; off >>= 1) acc += __shfl_xor(acc, off, 32);
    if (lane == 0) {
      const float y = s_mask[(size_t)b * NN + n] ? y1 : y0;
      const float m = (1.0f - (float)adj[(size_t)b * NN + n]) * 1e30f;
      sw[n] = acc + cb + y - m;
    }
  }
  __syncthreads();

  // ---- softmax over 512 scores (one score per thread) ----
  const float a = sw[tid];
  float m = a;
#pragma unroll
  for (int off = 16; off > 0; off >>= 1) m = fmaxf(m, __shfl_xor(m, off, 32));
  if (lane == 0) wred[wave] = m;
  __syncthreads();
  if (tid < 16) {
    float mm = wred[tid];
#pragma unroll
    for (int off = 8; off > 0; off >>= 1) mm = fmaxf(mm, __shfl_xor(mm, off, 16));
    if (tid == 0) wred[0] = mm;
  }
  __syncthreads();
  m = wred[0];
  __syncthreads();

  const float e = __expf(a - m);
  float s = e;
#pragma unroll
  for (int off = 16; off > 0; off >>= 1) s += __shfl_xor(s, off, 32);
  if (lane == 0) wred[wave] = s;
  __syncthreads();
  if (tid < 16) {
    float ss = wred[tid];
#pragma unroll
    for (int off = 8; off > 0; off >>= 1) ss += __shfl_xor(ss, off, 16);
    if (tid == 0) wred[0] = ss;
  }
  __syncthreads();
  const float w = e * (1.0f / wred[0]);
  sw[tid] = w;
  out_w[(size_t)b * NN + tid] = w;       // attn_weight (B,1,N) flat
  __syncthreads();

  // ---- attn_sum[d] = sum_n w[n] * V[b,n,d] ----
  const float* Vb = V + (size_t)b * NN * DD;
  const int d    = tid & (DD - 1);       // 0..255
  const int half = tid >> 8;             // 0 or 1
  float acc2 = 0.0f;
  const int nbeg = half * 256;
#pragma unroll 4
  for (int n = nbeg; n < nbeg + 256; ++n)
    acc2 += sw[n] * Vb[(size_t)n * DD + d];    // coalesced 1KB per half-block

  __syncthreads();
  sw[tid] = acc2;                         // reuse sw for half-combine
  __syncthreads();
  if (tid < DD) out_s[(size_t)b * DD + tid] = sw[tid] + sw[tid + DD];
}

// ---------------------------------------------------------------------------
extern "C" void kernel_launch(void* const* d_in, const int* in_sizes, int n_in,
                              void* d_out, int out_size, void* d_ws, size_t ws_size,
                              hipStream_t stream) {
  (void)in_sizes; (void)n_in; (void)out_size; (void)ws_size;

  const float* Q      = (const float*)d_in[0];
  const float* K      = (const float*)d_in[1];
  const float* V      = (const float*)d_in[2];
  const int*   adj    = (const int*)d_in[3];
  const int*   s_mask = (const int*)d_in[4];
  const float* W1     = (const float*)d_in[5];
  const float* b1     = (const float*)d_in[6];
  const float* W2     = (const float*)d_in[7];
  const float* b2     = (const float*)d_in[8];
  const float* W3     = (const float*)d_in[9];
  const float* b3     = (const float*)d_in[10];
  const float* rel    = (const float*)d_in[11];

  float* out_w = (float*)d_out;            // B*N
  float* out_s = out_w + (size_t)BB * NN;  // B*D

  float* q_ws = (float*)d_ws;                      // B*D
  float* p_ws = q_ws + (size_t)BB * DD;            // B*D
  float* y_ws = p_ws + (size_t)BB * DD;            // 2
  float* c_ws = y_ws + 2;                          // B

  // q = Q @ W1ᵀ + b1   (fp32 WMMA)
  wmma_gemm_f32<true, true><<<dim3(DD / 16, BB / 16), 32, 0, stream>>>(
      Q, W1, b1, q_ws, BB, DD, DD, DD, DD, DD);
  // p = q @ W2          (fp32 WMMA)
  wmma_gemm_f32<false, false><<<dim3(DD / 16, BB / 16), 32, 0, stream>>>(
      q_ws, W2, nullptr, p_ws, BB, DD, DD, DD, DD, DD);
  // y0/y1 scalars and c[b] = q[b]·b2
  y_precompute<<<1, DD, 0, stream>>>(rel, W3, b3, y_ws);
  qdotb2<<<BB, DD, 0, stream>>>(q_ws, b2, c_ws);
  // bandwidth-bound attention core
  gat_attention<<<BB, 512, 0, stream>>>(K, V, adj, s_mask, p_ws, y_ws, c_ws,
                                        out_w, out_s);
}